// LayerSet_66005057405258
// MI455X (gfx1250) — compile-verified
//
#include <hip/hip_runtime.h>
#include <hip/hip_bf16.h>

typedef __attribute__((ext_vector_type(2))) float v2f;
typedef __attribute__((ext_vector_type(8))) float v8f;

// Native FP32 atomic add (global_atomic_add_f32, non-returning form).
__device__ __forceinline__ void atomic_add_f32(float* p, float v) {
    unsafeAtomicAdd(p, v);
}

// ---------------------------------------------------------------------------
// Scratch zeroing
// ---------------------------------------------------------------------------
__global__ void zero_f32(float* __restrict__ p, long long n) {
    long long t = (long long)blockIdx.x * blockDim.x + threadIdx.x;
    if (t < n) p[t] = 0.0f;
}

// ---------------------------------------------------------------------------
// Phase A1: unique-edge scatter.  128 threads per edge (one element each).
//   agg_t[row]  += z[col]          (128 floats)
//   agg_tf[row] += tf[e]           (64 floats)
//   deg[row]    += 1
// ---------------------------------------------------------------------------
__global__ void scatter_unique(const float* __restrict__ z,
                               const float* __restrict__ tf,
                               const int* __restrict__ ue,
                               float* __restrict__ agg_t,
                               float* __restrict__ agg_tf,
                               float* __restrict__ deg,
                               long long EU) {
    long long t = (long long)blockIdx.x * 256 + threadIdx.x;
    if (t >= EU * 128LL) return;
    int e = (int)(t >> 7);
    int j = (int)(t & 127);
    int row = ue[e];
    int col = ue[EU + e];
    atomic_add_f32(&agg_t[(long long)row * 128 + j], z[(long long)col * 128 + j]);
    if (j < 64)
        atomic_add_f32(&agg_tf[(long long)row * 64 + j], tf[(long long)e * 64 + j]);
    if (j == 0)
        atomic_add_f32(&deg[row], 1.0f);
}

// ---------------------------------------------------------------------------
// Phase A2: time-diff segment sum over all E temporal edges.
// ---------------------------------------------------------------------------
__global__ void tsum_kernel(const int* __restrict__ ei,
                            const float* __restrict__ td,
                            float* __restrict__ tsum,
                            long long E) {
    long long t = (long long)blockIdx.x * 256 + threadIdx.x;
    if (t >= E) return;
    atomic_add_f32(&tsum[ei[t]], td[t]);
}

// ---------------------------------------------------------------------------
// Phase A3: weighted scatter.  w = td/tsum[row]
//   agg_s[row] += w * z[col];  relsum[row] += w
// ---------------------------------------------------------------------------
__global__ void scatter_weighted(const float* __restrict__ z,
                                 const int* __restrict__ ei,
                                 const float* __restrict__ td,
                                 const float* __restrict__ tsum,
                                 float* __restrict__ agg_s,
                                 float* __restrict__ relsum,
                                 long long E) {
    long long t = (long long)blockIdx.x * 256 + threadIdx.x;
    if (t >= E * 128LL) return;
    int e = (int)(t >> 7);
    int j = (int)(t & 127);
    int row = ei[e];
    int col = ei[E + e];
    float ts = tsum[row];
    ts = (ts == 0.0f) ? 1.0f : ts;
    float w = td[e] / ts;
    atomic_add_f32(&agg_s[(long long)row * 128 + j], w * z[(long long)col * 128 + j]);
    if (j == 0)
        atomic_add_f32(&relsum[row], w);
}

// ---------------------------------------------------------------------------
// Phase B: fused dense node update using V_WMMA_F32_16X16X4_F32.
// One block of 256 threads (8 wave32) handles 16 nodes; each wave owns a
// 16-column slice of the 128-wide hidden/output dimension.
//
// WMMA f32 16x16x4 operand mapping (wave32):
//   A (16x4):  lane L holds row M=L%16, K = {0,1} (L<16) or {2,3} (L>=16)
//   B (4x16):  lane L holds col N=L%16, K = {0,1} (L<16) or {2,3} (L>=16)
//   C/D:       VGPR i: lanes 0-15 -> (M=i, N=lane), lanes 16-31 -> (M=8+i)
// ---------------------------------------------------------------------------
__device__ __forceinline__ v8f wmma_k4(v8f acc, v2f a, v2f b) {
    return __builtin_amdgcn_wmma_f32_16x16x4_f32(false, a, false, b,
                                                 (short)0, acc, false, false);
}

// A tile in LDS, row-major [16][lda]; B in global row-major [K][128].
__device__ __forceinline__ v8f mma_loop(v8f acc,
                                        const float* __restrict__ lsA, int lda,
                                        const float* __restrict__ gB,
                                        int K, int n0, int lane) {
    const int mn   = lane & 15;
    const int koff = (lane >> 4) << 1;   // 0 for lanes 0-15, 2 for lanes 16-31
    for (int k = 0; k < K; k += 4) {
        v2f a, b;
        a.x = lsA[mn * lda + k + koff];
        a.y = lsA[mn * lda + k + koff + 1];
        b.x = gB[(k + koff) * 128 + n0 + mn];
        b.y = gB[(k + koff + 1) * 128 + n0 + mn];
        acc = wmma_k4(acc, a, b);
    }
    return acc;
}

__global__ __launch_bounds__(256) void fused_node_kernel(
    const float* __restrict__ z,
    const float* __restrict__ agg_t, const float* __restrict__ agg_tf,
    const float* __restrict__ agg_s,
    const float* __restrict__ deg,   const float* __restrict__ relsum,
    const float* __restrict__ WS_t,  const float* __restrict__ bS_t,
    const float* __restrict__ WT,    const float* __restrict__ bT,
    const float* __restrict__ Wtm,   const float* __restrict__ btm,
    const float* __restrict__ WS_s,  const float* __restrict__ bS_s,
    const float* __restrict__ WN,    const float* __restrict__ bN,
    const float* __restrict__ W1,    const float* __restrict__ b1,
    const float* __restrict__ W2,    const float* __restrict__ b2,
    float* __restrict__ out, int N) {

    __shared__ float ls_z[16 * 128];
    __shared__ float ls_t[16 * 128];
    __shared__ float ls_tf[16 * 64];
    __shared__ float ls_s[16 * 128];
    __shared__ float ls_c[16 * 256];   // [r_u | gamma_u]
    __shared__ float ls_h[16 * 128];   // relu(combined@W1 + b1)
    __shared__ float ls_deg[16];
    __shared__ float ls_rs[16];

    const int m0  = blockIdx.x * 16;
    const int tid = threadIdx.x;

    for (int i = tid; i < 16 * 128; i += 256) {
        int m = i >> 7, c = i & 127;
        bool ok = (m0 + m) < N;
        long long g = (long long)(m0 + m) * 128 + c;
        ls_z[i] = ok ? z[g]     : 0.0f;
        ls_t[i] = ok ? agg_t[g] : 0.0f;
        ls_s[i] = ok ? agg_s[g] : 0.0f;
    }
    for (int i = tid; i < 16 * 64; i += 256) {
        int m = i >> 6, c = i & 63;
        bool ok = (m0 + m) < N;
        ls_tf[i] = ok ? agg_tf[(long long)(m0 + m) * 64 + c] : 0.0f;
    }
    if (tid < 16) {
        bool ok = (m0 + tid) < N;
        ls_deg[tid] = ok ? deg[m0 + tid]    : 0.0f;
        ls_rs[tid]  = ok ? relsum[m0 + tid] : 0.0f;
    }
    __syncthreads();

    const int wave = tid >> 5;
    const int lane = tid & 31;
    const int n0   = wave * 16;
    const int nn   = n0 + (lane & 15);

    // ---- r_u = relu(z@WS_t + agg_t@WT + agg_tf@Wtm + bS_t + deg*(bT+btm))
    v8f accr;
    {
        float bs = bS_t[nn];
        float bt = bT[nn] + btm[nn];
#pragma unroll
        for (int i = 0; i < 8; ++i) {
            int m = (lane < 16) ? i : (8 + i);
            accr[i] = bs + ls_deg[m] * bt;
        }
    }
    accr = mma_loop(accr, ls_z, 128, WS_t, 128, n0, lane);
    accr = mma_loop(accr, ls_t, 128, WT, 128, n0, lane);
    accr = mma_loop(accr, ls_tf, 64, Wtm, 64, n0, lane);

    // ---- gamma_u = relu(z@WS_s + agg_s@WN + bS_s + relsum*bN)
    v8f accg;
    {
        float bs = bS_s[nn];
        float bn = bN[nn];
#pragma unroll
        for (int i = 0; i < 8; ++i) {
            int m = (lane < 16) ? i : (8 + i);
            accg[i] = bs + ls_rs[m] * bn;
        }
    }
    accg = mma_loop(accg, ls_z, 128, WS_s, 128, n0, lane);
    accg = mma_loop(accg, ls_s, 128, WN, 128, n0, lane);

#pragma unroll
    for (int i = 0; i < 8; ++i) {
        int m = (lane < 16) ? i : (8 + i);
        ls_c[m * 256 + nn]       = fmaxf(accr[i], 0.0f);
        ls_c[m * 256 + 128 + nn] = fmaxf(accg[i], 0.0f);
    }
    __syncthreads();

    // ---- h1 = relu(combined @ W1 + b1)     (K = 256)
    v8f acc1;
    {
        float b = b1[nn];
#pragma unroll
        for (int i = 0; i < 8; ++i) acc1[i] = b;
    }
    acc1 = mma_loop(acc1, ls_c, 256, W1, 256, n0, lane);
#pragma unroll
    for (int i = 0; i < 8; ++i) {
        int m = (lane < 16) ? i : (8 + i);
        ls_h[m * 128 + nn] = fmaxf(acc1[i], 0.0f);
    }
    __syncthreads();

    // ---- out = z + relu(h1 @ W2 + b2)      (K = 128)
    v8f acc2;
    {
        float b = b2[nn];
#pragma unroll
        for (int i = 0; i < 8; ++i) acc2[i] = b;
    }
    acc2 = mma_loop(acc2, ls_h, 128, W2, 128, n0, lane);
#pragma unroll
    for (int i = 0; i < 8; ++i) {
        int m = (lane < 16) ? i : (8 + i);
        if (m0 + m < N)
            out[(long long)(m0 + m) * 128 + nn] =
                ls_z[m * 128 + nn] + fmaxf(acc2[i], 0.0f);
    }
}

// ---------------------------------------------------------------------------
extern "C" void kernel_launch(void* const* d_in, const int* in_sizes, int n_in,
                              void* d_out, int out_size, void* d_ws, size_t ws_size,
                              hipStream_t stream) {
    const float* z    = (const float*)d_in[0];
    const int*   ei   = (const int*)  d_in[1];
    const float* tf   = (const float*)d_in[2];
    const float* td   = (const float*)d_in[3];
    const int*   ue   = (const int*)  d_in[4];
    const float* WS_t = (const float*)d_in[5];
    const float* bS_t = (const float*)d_in[6];
    const float* WT   = (const float*)d_in[7];
    const float* bT   = (const float*)d_in[8];
    const float* Wtm  = (const float*)d_in[9];
    const float* btm  = (const float*)d_in[10];
    const float* WS_s = (const float*)d_in[11];
    const float* bS_s = (const float*)d_in[12];
    const float* WN   = (const float*)d_in[13];
    const float* bN   = (const float*)d_in[14];
    const float* W1   = (const float*)d_in[15];
    const float* b1   = (const float*)d_in[16];
    const float* W2   = (const float*)d_in[17];
    const float* b2   = (const float*)d_in[18];
    float* out = (float*)d_out;

    const long long N  = in_sizes[0] / 128;   // 50000
    const long long E  = in_sizes[1] / 2;     // 800000
    const long long EU = in_sizes[4] / 2;     // 400000

    // Workspace layout (floats)
    float* agg_t  = (float*)d_ws;             // N*128
    float* agg_s  = agg_t  + N * 128;         // N*128
    float* agg_tf = agg_s  + N * 128;         // N*64
    float* deg    = agg_tf + N * 64;          // N
    float* tsum   = deg    + N;               // N
    float* rels   = tsum   + N;               // N
    const long long zf = N * (128 + 128 + 64 + 3);

    // 1) zero accumulators
    {
        long long blocks = (zf + 255) / 256;
        zero_f32<<<(unsigned)blocks, 256, 0, stream>>>((float*)d_ws, zf);
    }
    // 2) unique-edge scatter (agg_t, agg_tf, deg)
    {
        long long blocks = (EU * 128 + 255) / 256;
        scatter_unique<<<(unsigned)blocks, 256, 0, stream>>>(z, tf, ue, agg_t,
                                                             agg_tf, deg, EU);
    }
    // 3) time-diff segment sum
    {
        long long blocks = (E + 255) / 256;
        tsum_kernel<<<(unsigned)blocks, 256, 0, stream>>>(ei, td, tsum, E);
    }
    // 4) weighted scatter (agg_s, relsum) — depends on tsum (stream order)
    {
        long long blocks = (E * 128 + 255) / 256;
        scatter_weighted<<<(unsigned)blocks, 256, 0, stream>>>(z, ei, td, tsum,
                                                               agg_s, rels, E);
    }
    // 5) fused WMMA node update
    {
        long long blocks = (N + 15) / 16;
        fused_node_kernel<<<(unsigned)blocks, 256, 0, stream>>>(
            z, agg_t, agg_tf, agg_s, deg, rels,
            WS_t, bS_t, WT, bT, Wtm, btm,
            WS_s, bS_s, WN, bN, W1, b1, W2, b2,
            out, (int)N);
    }
}